// RadialMasking_53876069761147
// MI455X (gfx1250) — compile-verified
//
#include <hip/hip_runtime.h>
#include <math.h>

// ---------------- constants from the reference ----------------
#define RM_NGROUP 360
#define RM_NBATCH 4
#define RM_C      64
// keep probs: stage1 = 1-0.8 = 0.2 ; stage2 bins = {0.52, 0.20, 0.05, 0.0}

typedef float v2f __attribute__((ext_vector_type(2)));
typedef float v8f __attribute__((ext_vector_type(8)));

// ws header layout (ints):
//  [0,1440)      counts1   (batch*360 + group)
//  [1440,1444)   nb        (per-batch element count)
//  [1444,1448)   maxd bits (per-batch max kept distance, f32 bits via uint max)
//  [1448,1464)   counts2   (batch*4 + bin)
//  [1464,1468)   cnt_final (per-batch kept after stage2)
#define HDR_COUNTS1 0
#define HDR_NB      1440
#define HDR_MAXD    1444
#define HDR_COUNTS2 1448
#define HDR_CNTFIN  1464
#define HDR_INTS    2048

__device__ __forceinline__ float rm_rnd01(unsigned i, unsigned salt) {
    unsigned x = i * 0x9E3779B9u + salt * 0x85EBCA6Bu + 0x1234567u;
    x ^= x >> 16; x *= 0x7FEB352Du;
    x ^= x >> 15; x *= 0x846CA68Bu;
    x ^= x >> 16;
    return (float)(x >> 8) * (1.0f / 16777216.0f);
}

__device__ __forceinline__ void rm_geom(const int4* vc, int i,
                                        int& b, float& d, int& g) {
    int4 v = vc[i];                 // (b, z, y, x)
    b = v.x;
    float rx = -70.0f + (float)v.w * 0.1f;
    float ry = -40.0f + (float)v.z * 0.1f;
    d = sqrtf(rx * rx + ry * ry);
    float deg = atan2f(ry, rx) * 57.295779513082f;
    deg = deg - floorf(deg * (1.0f / 360.0f)) * 360.0f;   // mod into [0,360)
    g = (int)deg;
    if (g < 0) g = 0;
    if (g > RM_NGROUP - 1) g = RM_NGROUP - 1;
}

// ---------------- pass 0: zero counters ----------------
__global__ void rm_zero_hdr(int* hdr) {
    int i = blockIdx.x * blockDim.x + threadIdx.x;
    if (i < HDR_INTS) hdr[i] = 0;
}

// ---------------- pass 1: segment counts (stage 1) ----------------
__global__ void rm_count1(const int4* __restrict__ vc, int* __restrict__ hdr, int n) {
    int i = blockIdx.x * blockDim.x + threadIdx.x;
    if (i >= n) return;
    int b, g; float d;
    rm_geom(vc, i, b, d, g);
    atomicAdd(&hdr[HDR_COUNTS1 + b * RM_NGROUP + g], 1);
    atomicAdd(&hdr[HDR_NB + b], 1);
}

// ---------------- pass 2: keep1 decision + per-batch max kept d ----------------
__global__ void rm_keep1(const int4* __restrict__ vc, int* __restrict__ hdr,
                         unsigned char* __restrict__ keepbuf, int n) {
    int i = blockIdx.x * blockDim.x + threadIdx.x;
    if (i >= n) return;
    int b, g; float d;
    rm_geom(vc, i, b, d, g);
    int cnt = hdr[HDR_COUNTS1 + b * RM_NGROUP + g];
    int nk = (cnt > 0) ? max(1, (int)((float)cnt * 0.2f)) : 0;
    float u = rm_rnd01((unsigned)i, 1u);
    bool keep = (u * (float)cnt) < (float)nk;
    keepbuf[i] = keep ? 1 : 0;
    if (keep)
        atomicMax((unsigned int*)&hdr[HDR_MAXD + b], __float_as_uint(d));
}

// ---------------- pass 3: distance-bin counts among kept ----------------
__device__ __forceinline__ int rm_bin(float d, float m) {
    if (d < 0.3f * m) return 0;
    if (d < 0.7f * m) return 1;
    if (d < m)        return 2;
    return 3;
}

__global__ void rm_count2(const int4* __restrict__ vc, int* __restrict__ hdr,
                          const unsigned char* __restrict__ keepbuf, int n) {
    int i = blockIdx.x * blockDim.x + threadIdx.x;
    if (i >= n) return;
    if (!keepbuf[i]) return;
    int b, g; float d;
    rm_geom(vc, i, b, d, g);
    float m = __uint_as_float(((const unsigned int*)&hdr[HDR_MAXD])[b]);
    atomicAdd(&hdr[HDR_COUNTS2 + b * 4 + rm_bin(d, m)], 1);
}

// ---------------- pass 4: keep2 decision ----------------
__global__ void rm_keep2(const int4* __restrict__ vc, int* __restrict__ hdr,
                         unsigned char* __restrict__ keepbuf, int n) {
    int i = blockIdx.x * blockDim.x + threadIdx.x;
    if (i >= n) return;
    if (!keepbuf[i]) return;
    int b, g; float d;
    rm_geom(vc, i, b, d, g);
    float m = __uint_as_float(((const unsigned int*)&hdr[HDR_MAXD])[b]);
    int bin = rm_bin(d, m);
    const float p2tab[4] = {0.52f, 0.20f, 0.05f, 0.0f};
    float p = p2tab[bin];
    int cnt = hdr[HDR_COUNTS2 + b * 4 + bin];
    int nk = (cnt > 0 && p > 0.0f) ? max(1, (int)((float)cnt * p)) : 0;
    float u = rm_rnd01((unsigned)i, 2u);
    bool keep = (u * (float)cnt) < (float)nk;
    keepbuf[i] = keep ? 1 : 0;
    if (keep) atomicAdd(&hdr[HDR_CNTFIN + b], 1);
}

// ---------------- pass 5: fallback union, emit keep/scale/coords ----------------
__global__ void rm_finalize(const int* __restrict__ coords, const int* __restrict__ hdr,
                            const unsigned char* __restrict__ keepbuf,
                            float* __restrict__ scale,
                            float* __restrict__ out_coords,
                            float* __restrict__ out_keep, int n) {
    int i = blockIdx.x * blockDim.x + threadIdx.x;
    if (i >= n) return;
    int b = coords[4 * i];
    int nb = hdr[HDR_NB + b];
    int minv = max(1, (int)((float)nb * 0.05f));
    bool need = hdr[HDR_CNTFIN + b] < minv;
    float u = rm_rnd01((unsigned)i, 3u);
    bool extra = (u * (float)nb) < (float)minv;
    bool keep = (keepbuf[i] != 0) || (need && extra);
    float s = keep ? 1.0f : 0.0f;
    scale[i] = s;
    out_keep[i] = s;
    out_coords[4 * i + 0] = (float)coords[4 * i + 0];
    out_coords[4 * i + 1] = (float)coords[4 * i + 1];
    out_coords[4 * i + 2] = (float)coords[4 * i + 2];
    out_coords[4 * i + 3] = (float)coords[4 * i + 3];
}

// ---------------- pass 6: masked_features = diag(keep) x F via WMMA ----------------
// One wave handles 16 rows x 64 channels. Per 16x16 tile: D = sum_j A_j x B_j
// with A_j a 16x4 slice of diag(scale) (rows 4j..4j+3) and B_j the matching
// 4x16 slice of F. A 16x4 f32 layout (ISA 7.12.2): lanes 0-15 hold K=0,1;
// lanes 16-31 hold K=2,3. C/D 16x16 f32: VGPR i -> row i (lanes 0-15) / row
// i+8 (lanes 16-31), col = lane%16.
__global__ void rm_mask_wmma(const float* __restrict__ feat,
                             const float* __restrict__ scale,
                             float* __restrict__ outf, int n_rows) {
    int wave = blockIdx.x * (blockDim.x >> 5) + (threadIdx.x >> 5);
    int lane = threadIdx.x & 31;
    int r0 = wave * 16;
    if (r0 >= n_rows) return;          // wave-uniform: EXEC stays all-1s

    int m    = lane & 15;
    int half = lane >> 4;              // 0: K/row pair {0,1}, 1: {2,3}
    float sc = scale[r0 + m];
    int k0 = 2 * half;

    v2f A[4];
#pragma unroll
    for (int j = 0; j < 4; ++j) {
        A[j].x = (m == 4 * j + k0    ) ? sc : 0.0f;
        A[j].y = (m == 4 * j + k0 + 1) ? sc : 0.0f;
    }

    const float* base  = feat + (size_t)r0 * RM_C;
    float*       obase = outf + (size_t)r0 * RM_C;
    __builtin_prefetch(base + 16 * RM_C, 0, 1);   // next 16-row block

#pragma unroll
    for (int t = 0; t < 4; ++t) {
        int c0 = t * 16;
        v8f acc = {};
#pragma unroll
        for (int j = 0; j < 4; ++j) {
            int rA = 4 * j + 2 * half;             // this lane's B rows: rA, rA+1
            v2f B;
            B.x = base[(size_t)rA       * RM_C + c0 + m];
            B.y = base[(size_t)(rA + 1) * RM_C + c0 + m];
            acc = __builtin_amdgcn_wmma_f32_16x16x4_f32(
                false, A[j], false, B, (short)0, acc, false, false);
        }
#pragma unroll
        for (int i = 0; i < 8; ++i)
            obase[(size_t)(i + 8 * half) * RM_C + c0 + m] = acc[i];
    }
}

extern "C" void kernel_launch(void* const* d_in, const int* in_sizes, int n_in,
                              void* d_out, int out_size, void* d_ws, size_t ws_size,
                              hipStream_t stream) {
    const int*   coords = (const int*)d_in[0];
    const float* feat   = (const float*)d_in[1];
    const int N = in_sizes[0] / 4;

    float* out        = (float*)d_out;
    float* out_coords = out;                         // N*4
    float* out_feat   = out + (size_t)4 * N;         // N*64
    float* out_keep   = out + (size_t)68 * N;        // N

    char* ws = (char*)d_ws;
    int* hdr = (int*)ws;
    unsigned char* keepbuf = (unsigned char*)(ws + 8192);
    float* scale = (float*)(ws + 8192 + (((size_t)N + 255) & ~(size_t)255));

    const int TB = 256;
    const int GB = (N + TB - 1) / TB;

    rm_zero_hdr<<<(HDR_INTS + TB - 1) / TB, TB, 0, stream>>>(hdr);
    rm_count1  <<<GB, TB, 0, stream>>>((const int4*)coords, hdr, N);
    rm_keep1   <<<GB, TB, 0, stream>>>((const int4*)coords, hdr, keepbuf, N);
    rm_count2  <<<GB, TB, 0, stream>>>((const int4*)coords, hdr, keepbuf, N);
    rm_keep2   <<<GB, TB, 0, stream>>>((const int4*)coords, hdr, keepbuf, N);
    rm_finalize<<<GB, TB, 0, stream>>>(coords, hdr, keepbuf, scale,
                                       out_coords, out_keep, N);

    const int waves   = (N + 15) / 16;       // 16 rows per wave
    const int wpb     = TB / 32;             // 8 waves per block
    const int blocks7 = (waves + wpb - 1) / wpb;
    rm_mask_wmma<<<blocks7, TB, 0, stream>>>(feat, scale, out_feat, N);
}